// InverseCovMatrixFromPositions_21689584845409
// MI455X (gfx1250) — compile-verified
//
#include <hip/hip_runtime.h>

#define NB   20    // neighbor count (matrix dim)
#define LDA  21    // covA leading dim (A-operand only: rows 0..31, cols 0..19)
#define LDB  33    // X / T leading dim (used as B-operand: cols up to 31 + pad)
#define WPB  4     // waves (matrices) per 128-thread block

// per-wave LDS slice layout (floats):
//   covA : [0      , 672 )   32 rows x LDA
//   Xm   : [672    , 1728)   32 rows x LDB
//   Tm   : [1728   , 2784)   32 rows x LDB
// padded to 2816 floats (= 704 float4) for branch-free vectorized zero-fill
#define COV_OFF 0
#define X_OFF   672
#define T_OFF   1728
#define SLICE4  704   // float4s per wave slice

typedef __attribute__((ext_vector_type(2))) float v2f;
typedef __attribute__((ext_vector_type(8))) float v8f;

// One 16x16 output tile of the padded 32x32x20 f32 GEMM via V_WMMA_F32_16X16X4_F32.
// All padding is physically zero in LDS => no guards, loads merge into ds_load_2addr.
__device__ __forceinline__ v8f wmma_gemm_tile(const float* __restrict__ A, int lda,
                                              const float* __restrict__ B, int ldb,
                                              int mt, int nt, int lane) {
  v8f acc = {0.f, 0.f, 0.f, 0.f, 0.f, 0.f, 0.f, 0.f};
  const int ml = mt * 16 + (lane & 15);
  const int nl = nt * 16 + (lane & 15);
  const int kh = (lane < 16) ? 0 : 2;     // K half per lane group (ISA A/B layout)
#pragma unroll
  for (int kc = 0; kc < 5; ++kc) {        // K = 20 = 5 chunks of 4
    const int k0 = kc * 4 + kh;
    v2f a, b;
    a.x = A[ml * lda + k0];               // A 16x4: VGPR0=K(k0), VGPR1=K(k0+1)
    a.y = A[ml * lda + k0 + 1];
    b.x = B[k0 * ldb + nl];               // B 4x16: rows striped across lanes
    b.y = B[(k0 + 1) * ldb + nl];
    acc = __builtin_amdgcn_wmma_f32_16x16x4_f32(false, a, false, b,
                                                (short)0, acc, false, false);
  }
  return acc;
}

__global__ __launch_bounds__(128) void
InverseCovMatrixFromPositions_21689584845409_kernel(const float* __restrict__ pos,
                                                    float* __restrict__ out, int B) {
  const int lane = threadIdx.x & 31;
  const int wid  = threadIdx.x >> 5;
  const int b    = blockIdx.x * WPB + wid;

  __shared__ float4 lds4[WPB][SLICE4];
  float* base = reinterpret_cast<float*>(&lds4[wid][0]);
  float* covA = base + COV_OFF;
  float* Xm   = base + X_OFF;
  float* Tm   = base + T_OFF;

  if (b >= B) return;                     // wave-uniform: EXEC all-ones inside

  // ---- branch-free zero-fill of the whole slice (establishes zero padding) ----
  {
    const float4 z = make_float4(0.f, 0.f, 0.f, 0.f);
    float4* zp = &lds4[wid][0];
#pragma unroll
    for (int i = 0; i < SLICE4 / 32; ++i) zp[i * 32 + lane] = z;   // 22 b128 stores
  }

  // ---- positions: lane j owns point j ----
  float x = 0.f, y = 0.f;
  if (lane < NB) {
    const float2 p = reinterpret_cast<const float2*>(pos + (size_t)b * (NB * 2))[lane];
    x = p.x; y = p.y;
  }

  const float sigma2 = 1.0f, phi = 0.5f, tau = 0.1f;

  // ---- build covariance: lane j holds column j in registers ----
  float col[NB];
#pragma unroll
  for (int i = 0; i < NB; ++i) {
    const float xi = __shfl(x, i);                 // uniform index -> v_readlane
    const float yi = __shfl(y, i);
    const float dx = x - xi, dy = y - yi;
    const float d  = sqrtf(fmaf(dx, dx, dy * dy));
    float c = sigma2 * __expf(-phi * d);
    if (i == lane) c += tau * sigma2;              // nugget on the diagonal
    col[i] = c;
  }

  // stash A row-major (rows 20..31 stay zero from the fill)
  if (lane < NB) {
#pragma unroll
    for (int i = 0; i < NB; ++i) covA[i * LDA + lane] = col[i];
  }

  // ---- in-place Gauss-Jordan inversion (SPD + nugget => no pivoting) ----
#pragma unroll
  for (int k = 0; k < NB; ++k) {
    float f[NB];
#pragma unroll
    for (int i = 0; i < NB; ++i) f[i] = __shfl(col[i], k);   // old column k
    const float invp = __builtin_amdgcn_rcpf(f[k]);
    const float rk   = (lane == k) ? invp : col[k] * invp;
#pragma unroll
    for (int i = 0; i < NB; ++i) {
      if (i == k) continue;
      const float ci = (lane == k) ? 0.0f : col[i];          // implicit identity col
      col[i] = fmaf(-f[i], rk, ci);
    }
    col[k] = rk;
  }

  // stash X = A^-1 row-major (pad rows/cols stay zero)
  if (lane < NB) {
#pragma unroll
    for (int i = 0; i < NB; ++i) Xm[i * LDB + lane] = col[i];
  }
  __asm__ volatile("s_wait_dscnt 0" ::: "memory");   // same-wave LDS produce->consume

  // ---- Newton-Schulz refinement, f32 WMMA:  T = A*X  (full tiles; pad self-zeroes) ----
#pragma unroll
  for (int mt = 0; mt < 2; ++mt)
#pragma unroll
    for (int nt = 0; nt < 2; ++nt) {
      const v8f t  = wmma_gemm_tile(covA, LDA, Xm, LDB, mt, nt, lane);
      const int nl = nt * 16 + (lane & 15);
#pragma unroll
      for (int v = 0; v < 8; ++v) {                  // C/D: VGPR v -> M = v / v+8
        const int ml = mt * 16 + v + ((lane < 16) ? 0 : 8);
        Tm[ml * LDB + nl] = t[v];                    // unconditional: zeros land in pad
      }
    }
  __asm__ volatile("s_wait_dscnt 0" ::: "memory");

  // ---- X1 = 2X - X*T, streamed straight to HBM ----
  float* o = out + (size_t)b * (NB * NB);
#pragma unroll
  for (int mt = 0; mt < 2; ++mt)
#pragma unroll
    for (int nt = 0; nt < 2; ++nt) {
      const v8f u  = wmma_gemm_tile(Xm, LDB, Tm, LDB, mt, nt, lane);
      const int nl = nt * 16 + (lane & 15);
#pragma unroll
      for (int v = 0; v < 8; ++v) {
        const int ml = mt * 16 + v + ((lane < 16) ? 0 : 8);
        if (ml < NB && nl < NB)
          o[ml * NB + nl] = fmaf(2.0f, Xm[ml * LDB + nl], -u[v]);
      }
    }
}

extern "C" void kernel_launch(void* const* d_in, const int* in_sizes, int n_in,
                              void* d_out, int out_size, void* d_ws, size_t ws_size,
                              hipStream_t stream) {
  const float* pos = (const float*)d_in[0];
  // d_in[1] (edge_list) is unused by the reference computation.
  const int B = in_sizes[0] / (NB * 2);
  float* out = (float*)d_out;
  const int blocks = (B + WPB - 1) / WPB;
  hipLaunchKernelGGL(InverseCovMatrixFromPositions_21689584845409_kernel,
                     dim3(blocks), dim3(128), 0, stream, pos, out, B);
}